// MLP_Core_6442450944689
// MI455X (gfx1250) — compile-verified
//
#include <hip/hip_runtime.h>
#include <hip/hip_bf16.h>

#define H_DIM 4096
#define I_DIM 14336
#define HID   1024
#define SC_N  4300
#define HC_N  716
#define K_PAD 5120   // 40 blocks * 128 rows, >= 5016 max unique
#define PF_DIST 1024 // prefetch distance in floats (4KB ahead in the row)

typedef __attribute__((ext_vector_type(2))) float v2f;
typedef __attribute__((ext_vector_type(4))) float v4f;
typedef __attribute__((ext_vector_type(8))) float v8f;

__device__ __forceinline__ v8f wmma_f32_k4(v2f a, v2f b, v8f c) {
  // D = A(16x4 f32) * B(4x16 f32) + C ; emits v_wmma_f32_16x16x4_f32
  return __builtin_amdgcn_wmma_f32_16x16x4_f32(false, a, false, b, (short)0, c,
                                               false, false);
}

__device__ __forceinline__ v2f nt_load2(const float* p) {
  // streaming (non-temporal) 8B load: weights are single-use, keep L2 for
  // the reused small operands (x / h / tvdense / idx_list)
  return __builtin_nontemporal_load((const v2f*)p);
}

// ---------------- clear bitmap + dense tv vector (each call) ----------------
__global__ void k_clear(unsigned int* __restrict__ bitmap,
                        float* __restrict__ tvdense) {
  int t = blockIdx.x * blockDim.x + threadIdx.x;
  if (t < I_DIM / 32) bitmap[t] = 0u;
  if (t < I_DIM) tvdense[t] = 0.0f;
}

// ---------------- mark x_topk members in bitmap ----------------
__global__ void k_mark(const int* __restrict__ topk,
                       unsigned int* __restrict__ bitmap) {
  int t = blockIdx.x * blockDim.x + threadIdx.x;
  if (t < SC_N) {
    int v = topk[t];
    if (v >= 0 && v < I_DIM) atomicOr(&bitmap[v >> 5], 1u << (v & 31));
  }
}

// ---------------- generic WMMA GEMV: out[r] = dot(vec, W[r,:]) ----------------
// grid.x * 128 rows total; block = 256 threads (8 waves), 16 rows per wave.
// A is the broadcast vector chunk, B is a 4x16 tile of 16 weight rows.
__global__ void k_wmma_gemv(const float* __restrict__ W,
                            const float* __restrict__ vec,
                            float* __restrict__ out, int veclen) {
  extern __shared__ float sv[];
  for (int i = threadIdx.x; i < veclen; i += blockDim.x) sv[i] = vec[i];
  __syncthreads();

  const int lane = threadIdx.x & 31;
  const int wave = threadIdx.x >> 5;
  const int n    = lane & 15;       // output row within tile (B N-dim)
  const int half = lane >> 4;       // K pair select: 0 -> K0,K1 ; 1 -> K2,K3
  const int rowBase = (blockIdx.x * 8 + wave) * 16;

  const float* wr = W + (size_t)(rowBase + n) * (size_t)veclen + 2 * half;
  const float* xv = sv + 2 * half;

  v8f acc0 = {}, acc1 = {};  // two chains -> independent WMMA issue
  for (int h0 = 0; h0 < veclen; h0 += 32) {
    __builtin_prefetch(wr + h0 + PF_DIST, 0, 0);  // global_prefetch_b8
#pragma unroll
    for (int u = 0; u < 4; ++u) {
      const int hh = h0 + u * 8;
      v2f a0 = *(const v2f*)(xv + hh);
      v2f a1 = *(const v2f*)(xv + hh + 4);
      v2f b0 = nt_load2(wr + hh);
      v2f b1 = nt_load2(wr + hh + 4);
      acc0 = wmma_f32_k4(a0, b0, acc0);
      acc1 = wmma_f32_k4(a1, b1, acc1);
    }
  }
  v8f acc = acc0 + acc1;
  // D rows identical (A replicated); M=0 row lives in acc[0] of lanes 0..15
  if (lane < 16) out[rowBase + lane] = acc[0];
}

// ---------------- exact top-HC by rank counting; mark bitmap ----------------
__global__ void k_topk_mark(const float* __restrict__ h,
                            unsigned int* __restrict__ bitmap) {
  extern __shared__ float sh[];
  for (int j = threadIdx.x; j < I_DIM; j += blockDim.x) sh[j] = h[j];
  __syncthreads();
  const int i = blockIdx.x * blockDim.x + threadIdx.x;  // 56*256 == I_DIM
  const float hi = sh[i];
  int cnt = 0;
  const v4f* s4 = (const v4f*)sh;
  for (int jj = 0; jj < I_DIM / 4; ++jj) {
    v4f v = s4[jj];
    int j = jj * 4;
    cnt += (v[0] > hi) || (v[0] == hi && (j + 0) < i);
    cnt += (v[1] > hi) || (v[1] == hi && (j + 1) < i);
    cnt += (v[2] > hi) || (v[2] == hi && (j + 2) < i);
    cnt += (v[3] > hi) || (v[3] == hi && (j + 3) < i);
  }
  if (cnt < HC_N) atomicOr(&bitmap[i >> 5], 1u << (i & 31));
}

// ---------------- bitmap -> sorted unique index list (== jnp.unique) --------
__global__ void k_compact(const unsigned int* __restrict__ bitmap,
                          int* __restrict__ idx_list,
                          int* __restrict__ kcount) {
  __shared__ unsigned int s[512];
  const int t = threadIdx.x;
  unsigned int w = (t < I_DIM / 32) ? bitmap[t] : 0u;
  unsigned int c = __popc(w);
  s[t] = c;
  __syncthreads();
  for (int off = 1; off < 512; off <<= 1) {
    unsigned int v = (t >= off) ? s[t - off] : 0u;
    __syncthreads();
    s[t] += v;
    __syncthreads();
  }
  unsigned int pos = s[t] - c;  // exclusive prefix
  unsigned int ww = w;
  while (ww) {
    int b = __ffs((int)ww) - 1;
    idx_list[pos++] = (t << 5) + b;
    ww &= ww - 1;
  }
  if (t == 511) *kcount = (int)s[511];
}

// ---------------- gather gate/up rows, WMMA GEMV, silu, scatter tv ----------
__global__ void k_gateup(const float* __restrict__ gate_w,
                         const float* __restrict__ up_w,
                         const float* __restrict__ x,
                         const int* __restrict__ idx_list,
                         const int* __restrict__ kcount,
                         float* __restrict__ tvdense) {
  __shared__ float sx[H_DIM];
  for (int i = threadIdx.x; i < H_DIM; i += blockDim.x) sx[i] = x[i];
  __syncthreads();
  const int kc   = *kcount;
  const int lane = threadIdx.x & 31;
  const int wave = threadIdx.x >> 5;
  const int n    = lane & 15;
  const int half = lane >> 4;
  const int kBase = (blockIdx.x * 8 + wave) * 16;
  const int kk   = kBase + n;
  const int kcl  = (kk < kc) ? kk : (kc - 1);  // clamp keeps EXEC uniform
  const int row  = idx_list[kcl];

  const float* gr = gate_w + (size_t)row * H_DIM + 2 * half;
  const float* ur = up_w   + (size_t)row * H_DIM + 2 * half;
  const float* xv = sx + 2 * half;

  v8f ag0 = {}, ag1 = {}, au0 = {}, au1 = {};
  for (int h0 = 0; h0 < H_DIM; h0 += 32) {
    __builtin_prefetch(gr + h0 + PF_DIST, 0, 0);
    __builtin_prefetch(ur + h0 + PF_DIST, 0, 0);
#pragma unroll
    for (int u = 0; u < 4; ++u) {
      const int hh = h0 + u * 8;
      v2f a0  = *(const v2f*)(xv + hh);
      v2f a1  = *(const v2f*)(xv + hh + 4);
      v2f bg0 = nt_load2(gr + hh);
      v2f bg1 = nt_load2(gr + hh + 4);
      v2f bu0 = nt_load2(ur + hh);
      v2f bu1 = nt_load2(ur + hh + 4);
      ag0 = wmma_f32_k4(a0, bg0, ag0);
      au0 = wmma_f32_k4(a0, bu0, au0);
      ag1 = wmma_f32_k4(a1, bg1, ag1);
      au1 = wmma_f32_k4(a1, bu1, au1);
    }
  }
  v8f accg = ag0 + ag1;
  v8f accu = au0 + au1;
  if (lane < 16 && kk < kc) {
    float g = accg[0];
    float u = accu[0];
    float t = (g / (1.0f + __expf(-g))) * u;  // silu(g) * u
    tvdense[row] = t;                         // rows unique -> no conflicts
  }
}

extern "C" void kernel_launch(void* const* d_in, const int* in_sizes, int n_in,
                              void* d_out, int out_size, void* d_ws,
                              size_t ws_size, hipStream_t stream) {
  const float* x       = (const float*)d_in[0];
  const int*   x_topk  = (const int*)d_in[1];
  const float* gate_w  = (const float*)d_in[2];
  const float* up_w    = (const float*)d_in[3];
  const float* down_w  = (const float*)d_in[4];
  const float* hw1     = (const float*)d_in[5];
  const float* hw2     = (const float*)d_in[6];
  float*       out     = (float*)d_out;

  float* h1            = (float*)d_ws;          // 1024
  float* h             = h1 + HID;              // 14336
  float* tvd           = h + I_DIM;             // 14336
  unsigned int* bitmap = (unsigned int*)(tvd + I_DIM);  // 448
  int*   idx_list      = (int*)(bitmap + I_DIM / 32);   // 5120
  int*   kcount        = idx_list + K_PAD;              // 1

  // fresh state every call (graph-replay safe)
  k_clear<<<I_DIM / 256, 256, 0, stream>>>(bitmap, tvd);
  k_mark<<<(SC_N + 255) / 256, 256, 0, stream>>>(x_topk, bitmap);

  // h1 = helper_w1 @ x ; h = helper_w2 @ h1
  k_wmma_gemv<<<HID / 128, 256, H_DIM * 4, stream>>>(hw1, x, h1, H_DIM);
  k_wmma_gemv<<<I_DIM / 128, 256, HID * 4, stream>>>(hw2, h1, h, HID);

  // top-HC of h -> bitmap ; bitmap -> sorted unique idx list + count
  k_topk_mark<<<I_DIM / 256, 256, I_DIM * 4, stream>>>(h, bitmap);
  k_compact<<<1, 512, 0, stream>>>(bitmap, idx_list, kcount);

  // tv scattered dense, then dense down GEMV: out = down_w @ tvdense
  k_gateup<<<K_PAD / 128, 256, 0, stream>>>(gate_w, up_w, x, idx_list, kcount,
                                            tvd);
  k_wmma_gemv<<<H_DIM / 128, 256, I_DIM * 4, stream>>>(down_w, tvd, out,
                                                       I_DIM);
}